// GeometricVQ_57870389347068
// MI455X (gfx1250) — compile-verified
//
#include <hip/hip_runtime.h>

typedef __attribute__((ext_vector_type(2))) float v2f;
typedef __attribute__((ext_vector_type(8))) float v8f;

#define N_E   1024
#define DDIM  128
#define BATCH 16
#define NSEQ  4096
#define ROWS  (BATCH * NSEQ)     /* 65536 */
#define ZELEMS (ROWS * DDIM)     /* 8388608 */

/* d_out offsets (floats), in reference tuple return order */
#define OFF_ZQST 0
#define OFF_LOSS 8388608
#define OFF_SAMP 8388609
#define OFF_IDX  8404993
#define OFF_VAR  8470529
#define OFF_DIST 8470530
#define OFF_SI   8470531
#define OFF_ZN   8536067
/* total = 16924675 */

/* workspace offsets (float units) */
#define WS_WN2   0        /* 1024  */
#define WS_MIN2  1024     /* 1024  */
#define WS_VAR   2048     /* 1024  */
#define WS_IDX   3072     /* int[65536] */
#define WS_NPART 68608    /* 16*16*128  */
#define WS_NORM  101376   /* 2048  */
#define WS_LPART 103424   /* 4096  */

/* ---------------- trivial helpers ---------------- */

__global__ void k_zero_samp(float* out) {
  int i = blockIdx.x * 256 + threadIdx.x;
  if (i < BATCH * N_E) out[OFF_SAMP + i] = 0.0f;
}

__global__ void k_wn2(const float* __restrict__ w, float* __restrict__ wn2) {
  int j = blockIdx.x * 256 + threadIdx.x;
  if (j < N_E) {
    float s = 0.f;
    for (int k = 0; k < DDIM; ++k) { float v = w[j * DDIM + k]; s += v * v; }
    wn2[j] = s;
  }
}

/* ------------- codebook self-distance stats ------------- */
__global__ void k_cbstats(const float* __restrict__ w, const float* __restrict__ wn2,
                          float* __restrict__ rowmin2, float* __restrict__ rowvar) {
  __shared__ float wi[DDIM];
  __shared__ float cdrow[N_E];
  __shared__ float red[256];
  __shared__ float rm1[256];
  __shared__ float rm2[256];
  const int i = blockIdx.x, tid = threadIdx.x;
  if (tid < DDIM) wi[tid] = w[i * DDIM + tid];
  __syncthreads();
  const float w2i = wn2[i];
  for (int j = tid; j < N_E; j += 256) {
    const float* wj = w + j * DDIM;
    float dot = 0.f;
    for (int k = 0; k < DDIM; ++k) dot += wi[k] * wj[k];
    cdrow[j] = w2i + wn2[j] - 2.0f * dot;
  }
  __syncthreads();
  /* mean */
  float s = 0.f;
  for (int j = tid; j < N_E; j += 256) s += cdrow[j];
  red[tid] = s; __syncthreads();
  for (int off = 128; off > 0; off >>= 1) { if (tid < off) red[tid] += red[tid + off]; __syncthreads(); }
  const float mean = red[0] / (float)N_E;
  __syncthreads();
  /* unbiased variance (two-pass) */
  float sv = 0.f;
  for (int j = tid; j < N_E; j += 256) { float dd = cdrow[j] - mean; sv += dd * dd; }
  red[tid] = sv; __syncthreads();
  for (int off = 128; off > 0; off >>= 1) { if (tid < off) red[tid] += red[tid + off]; __syncthreads(); }
  /* two smallest -> kthvalue(.,2) */
  float m1 = 3.4e38f, m2 = 3.4e38f;
  for (int j = tid; j < N_E; j += 256) {
    float d = cdrow[j];
    if (d < m1) { m2 = m1; m1 = d; } else if (d < m2) { m2 = d; }
  }
  rm1[tid] = m1; rm2[tid] = m2; __syncthreads();
  for (int off = 128; off > 0; off >>= 1) {
    if (tid < off) {
      float a1 = rm1[tid], a2 = rm2[tid], b1 = rm1[tid + off], b2 = rm2[tid + off];
      rm1[tid] = fminf(a1, b1);
      rm2[tid] = fminf(fmaxf(a1, b1), fminf(a2, b2));
    }
    __syncthreads();
  }
  if (tid == 0) { rowmin2[i] = rm2[0]; rowvar[i] = red[0] / (float)(N_E - 1); }
}

__global__ void k_cbreduce(const float* __restrict__ rowmin2, const float* __restrict__ rowvar,
                           float* __restrict__ out) {
  __shared__ float r1[256];
  __shared__ float r2[256];
  const int tid = threadIdx.x;
  float s1 = 0.f, s2 = 0.f;
  for (int i = tid; i < N_E; i += 256) { s1 += rowmin2[i]; s2 += rowvar[i]; }
  r1[tid] = s1; r2[tid] = s2; __syncthreads();
  for (int off = 128; off > 0; off >>= 1) {
    if (tid < off) { r1[tid] += r1[tid + off]; r2[tid] += r2[tid + off]; }
    __syncthreads();
  }
  if (tid == 0) { out[OFF_DIST] = r1[0] / (float)N_E; out[OFF_VAR] = r2[0] / (float)N_E; }
}

/* ------------- main WMMA argmin kernel ------------- */
/* 8 waves/block; each wave owns 32 z-rows (two 16-row A tiles -> two independent
   WMMA accumulation chains). Codebook tiles double-buffered through LDS with
   register-staged global prefetch of tile t+1 under tile t's 64 WMMAs. */
__launch_bounds__(256)
__global__ void k_argmin(const float* __restrict__ z, const float* __restrict__ w,
                         const float* __restrict__ wn2, int* __restrict__ idxout,
                         float* __restrict__ out) {
  __shared__ float lw[2][16 * DDIM]; /* 2 x 8 KB tiles: 16 codes x 128 */
  const int tid   = threadIdx.x;
  const int lane  = tid & 31;
  const int wave  = tid >> 5;
  const int lmod  = lane & 15;
  const int lhalf = lane >> 4;
  const int mbase = (blockIdx.x * 8 + wave) * 32;

  /* A fragments: lane holds row m = mbase + lane%16 (+16 for set 1),
     K pair = kc*4 + 2*(lane/16) + {0,1} per the 16x4 f32 A layout */
  v2f a0[32], a1[32];
  const float* zrow0 = z + (mbase + lmod) * DDIM + 2 * lhalf;
  const float* zrow1 = zrow0 + 16 * DDIM;
#pragma unroll
  for (int kc = 0; kc < 32; ++kc) {
    a0[kc] = *(const v2f*)(zrow0 + kc * 4);
    a1[kc] = *(const v2f*)(zrow1 + kc * 4);
  }

  float bestd[16];
  int   besti[16];
#pragma unroll
  for (int r = 0; r < 16; ++r) { bestd[r] = 3.4e38f; besti[r] = 0; }

  /* prologue: register-stage tile 0 */
  float g[8];
#pragma unroll
  for (int j = 0; j < 8; ++j) g[j] = w[tid + j * 256];

  for (int t = 0; t < 64; ++t) {
    const int buf = t & 1;
#pragma unroll
    for (int j = 0; j < 8; ++j) lw[buf][tid + j * 256] = g[j];
    __syncthreads();
    if (t < 63) {
#pragma unroll
      for (int j = 0; j < 8; ++j) g[j] = w[(t + 1) * (16 * DDIM) + tid + j * 256];
    }

    /* B fragments: lane holds col n = t*16 + lane%16, same K pairing as A */
    v2f bfr[32];
    const float* wrow = &lw[buf][lmod * DDIM + 2 * lhalf];
#pragma unroll
    for (int kc = 0; kc < 32; ++kc) bfr[kc] = *(const v2f*)(wrow + kc * 4);

    /* keep the LDS loads issued ahead of the WMMA chain */
    asm volatile("" ::: "memory");

    v8f acc0 = {0.f, 0.f, 0.f, 0.f, 0.f, 0.f, 0.f, 0.f};
    v8f acc1 = {0.f, 0.f, 0.f, 0.f, 0.f, 0.f, 0.f, 0.f};
#pragma unroll
    for (int kc = 0; kc < 32; ++kc) {
      acc0 = __builtin_amdgcn_wmma_f32_16x16x4_f32(false, a0[kc], false, bfr[kc],
                                                   (short)0, acc0, false, false);
      acc1 = __builtin_amdgcn_wmma_f32_16x16x4_f32(false, a1[kc], false, bfr[kc],
                                                   (short)0, acc1, false, false);
    }

    const int n = t * 16 + lmod;
    const float w2 = wn2[n];
#pragma unroll
    for (int r = 0; r < 8; ++r) {
      float d0 = w2 - 2.0f * acc0[r];      /* ||z||^2 constant per row: dropped */
      if (d0 < bestd[r])     { bestd[r]     = d0; besti[r]     = n; }
      float d1 = w2 - 2.0f * acc1[r];
      if (d1 < bestd[8 + r]) { bestd[8 + r] = d1; besti[8 + r] = n; }
    }
  }

  /* argmin across the 16 lanes of each half (C/D layout: M = r + 8*(lane/16), N = lane%16) */
#pragma unroll
  for (int off = 1; off < 16; off <<= 1) {
#pragma unroll
    for (int r = 0; r < 16; ++r) {
      float od = __shfl_xor(bestd[r], off, 16);
      int   oi = __shfl_xor(besti[r], off, 16);
      if (od < bestd[r] || (od == bestd[r] && oi < besti[r])) { bestd[r] = od; besti[r] = oi; }
    }
  }
  if (lmod == 0) {
#pragma unroll
    for (int r = 0; r < 8; ++r) {
      int row0 = mbase + r + 8 * lhalf;
      int bi0  = besti[r];
      idxout[row0]        = bi0;
      out[OFF_IDX + row0] = (float)bi0;
      out[OFF_SI  + row0] = (float)bi0;
      out[OFF_SAMP + bi0] = 1.0f;   /* reference scatters flat idx in [0,1024): row 0 only */
      int row1 = mbase + 16 + r + 8 * lhalf;
      int bi1  = besti[8 + r];
      idxout[row1]        = bi1;
      out[OFF_IDX + row1] = (float)bi1;
      out[OFF_SI  + row1] = (float)bi1;
      out[OFF_SAMP + bi1] = 1.0f;
    }
  }
}

/* ------------- per-(b,d) norms over the N axis (F.normalize dim=1) ------------- */
__global__ void k_znormpart(const float* __restrict__ w, const int* __restrict__ idx,
                            float* __restrict__ part) {
  const int b = blockIdx.x;   /* 16 */
  const int c = blockIdx.y;   /* 16 chunks of 256 */
  const int d = threadIdx.x;  /* 128 */
  float s = 0.f;
  const int nbase = c * 256;
  for (int n = 0; n < 256; ++n) {
    int id = idx[b * NSEQ + nbase + n];
    float v = w[id * DDIM + d];
    s += v * v;
  }
  part[(b * 16 + c) * DDIM + d] = s;
}

__global__ void k_znormcombine(const float* __restrict__ part, float* __restrict__ norms) {
  int i = blockIdx.x * 256 + threadIdx.x; /* 2048 = 16*128 */
  if (i < BATCH * DDIM) {
    int b = i >> 7, d = i & 127;
    float s = 0.f;
    for (int c = 0; c < 16; ++c) s += part[(b * 16 + c) * DDIM + d];
    norms[i] = sqrtf(s);
  }
}

/* ------------- elementwise: z_q_st, zn, loss partials ------------- */
__global__ void k_ewise(const float* __restrict__ z, const float* __restrict__ w,
                        const int* __restrict__ idx, const float* __restrict__ norms,
                        float* __restrict__ out, float* __restrict__ lpart) {
  __shared__ float red[256];
  const int tid  = threadIdx.x;
  const int base = blockIdx.x * 2048;
  float acc = 0.f;
#pragma unroll
  for (int j = 0; j < 8; ++j) {
    int e   = base + j * 256 + tid;
    int row = e >> 7;
    int d   = e & 127;
    int b   = row >> 12;
    float zq = w[idx[row] * DDIM + d];
    float zv = z[e];
    float t  = zq - zv;                 /* sg(z_q) - z, same rounding as reference */
    out[OFF_ZQST + e] = zv + t;         /* z + sg(z_q - z) */
    acc += t * t;
    out[OFF_ZN + e] = zq / fmaxf(norms[(b << 7) + d], 1e-12f);
  }
  red[tid] = acc; __syncthreads();
  for (int off = 128; off > 0; off >>= 1) { if (tid < off) red[tid] += red[tid + off]; __syncthreads(); }
  if (tid == 0) lpart[blockIdx.x] = red[0];
}

__global__ void k_lossreduce(const float* __restrict__ lpart, float* __restrict__ out) {
  __shared__ float red[256];
  const int tid = threadIdx.x;
  float s = 0.f;
  for (int i = tid; i < 4096; i += 256) s += lpart[i];
  red[tid] = s; __syncthreads();
  for (int off = 128; off > 0; off >>= 1) { if (tid < off) red[tid] += red[tid + off]; __syncthreads(); }
  if (tid == 0) out[OFF_LOSS] = (1.0f + 0.9f) * red[0] / (float)ZELEMS;
}

/* ---------------- launch ---------------- */
extern "C" void kernel_launch(void* const* d_in, const int* in_sizes, int n_in,
                              void* d_out, int out_size, void* d_ws, size_t ws_size,
                              hipStream_t stream) {
  (void)in_sizes; (void)n_in; (void)out_size; (void)ws_size;
  const float* z = (const float*)d_in[0];
  const float* w = (const float*)d_in[1];
  float* out = (float*)d_out;
  float* ws  = (float*)d_ws;

  float* wn2     = ws + WS_WN2;
  float* rowmin2 = ws + WS_MIN2;
  float* rowvar  = ws + WS_VAR;
  int*   idxbuf  = (int*)(ws + WS_IDX);
  float* npart   = ws + WS_NPART;
  float* norms   = ws + WS_NORM;
  float* lpart   = ws + WS_LPART;

  k_zero_samp   <<<64,   256, 0, stream>>>(out);
  k_wn2         <<<4,    256, 0, stream>>>(w, wn2);
  k_cbstats     <<<N_E,  256, 0, stream>>>(w, wn2, rowmin2, rowvar);
  k_cbreduce    <<<1,    256, 0, stream>>>(rowmin2, rowvar, out);
  k_argmin      <<<256,  256, 0, stream>>>(z, w, wn2, idxbuf, out);
  k_znormpart   <<<dim3(16, 16), 128, 0, stream>>>(w, idxbuf, npart);
  k_znormcombine<<<8,    256, 0, stream>>>(npart, norms);
  k_ewise       <<<4096, 256, 0, stream>>>(z, w, idxbuf, norms, out, lpart);
  k_lossreduce  <<<1,    256, 0, stream>>>(lpart, out);
}